// SpatialTransformer_8607114461613
// MI455X (gfx1250) — compile-verified
//
#include <hip/hip_runtime.h>
#include <hip/hip_bf16.h>

typedef __attribute__((ext_vector_type(2))) float v2f;
typedef __attribute__((ext_vector_type(8))) float v8f;

#define NPTS   8192
#define KNN    5
#define EPSW   1e-8f
#define WAVES  8
#define CHUNK  64            // columns per wave per iteration
#define STAGE  (WAVES*CHUNK) // 512 columns staged per block iteration
#define CPAD   68            // LDS row stride: 272B = 16B-aligned rows, banks (4m+c)%64 conflict-free

// ---------------------------------------------------------------------------
// Kernel 1: localization MLP -> theta -> new_pos, plus squared norms.
// One thread per point; weights staged in LDS; loops fully unrolled so the
// hidden activations stay in VGPRs.
// ---------------------------------------------------------------------------
__global__ void st_mlp_kernel(const float* __restrict__ pos,
                              const float* __restrict__ W1, const float* __restrict__ b1,
                              const float* __restrict__ W2, const float* __restrict__ b2,
                              const float* __restrict__ W3, const float* __restrict__ b3,
                              float* __restrict__ newx, float* __restrict__ newy,
                              float* __restrict__ anorm, float* __restrict__ bnorm)
{
    __shared__ float sW1[64], sb1[32], sW2[2048], sb2[64], sW3[384], sb3[6];
    const int t = threadIdx.x;
    for (int k = t; k < 64;   k += blockDim.x) sW1[k] = W1[k];
    for (int k = t; k < 32;   k += blockDim.x) sb1[k] = b1[k];
    for (int k = t; k < 2048; k += blockDim.x) sW2[k] = W2[k];
    for (int k = t; k < 64;   k += blockDim.x) sb2[k] = b2[k];
    for (int k = t; k < 384;  k += blockDim.x) sW3[k] = W3[k];
    for (int k = t; k < 6;    k += blockDim.x) sb3[k] = b3[k];
    __syncthreads();

    const int i = blockIdx.x * blockDim.x + t;
    const float x = pos[2*i + 0];
    const float y = pos[2*i + 1];

    float h1[32];
#pragma unroll
    for (int o = 0; o < 32; ++o) {
        float acc = fmaf(sW1[2*o+1], y, fmaf(sW1[2*o], x, sb1[o]));
        h1[o] = fmaxf(acc, 0.0f);
    }
    float h2[64];
#pragma unroll
    for (int o = 0; o < 64; ++o) {
        float acc = sb2[o];
#pragma unroll
        for (int k = 0; k < 32; ++k) acc = fmaf(sW2[o*32 + k], h1[k], acc);
        h2[o] = fmaxf(acc, 0.0f);
    }
    float th[6];
#pragma unroll
    for (int o = 0; o < 6; ++o) {
        float acc = sb3[o];
#pragma unroll
        for (int k = 0; k < 64; ++k) acc = fmaf(sW3[o*64 + k], h2[k], acc);
        th[o] = acc;
    }
    const float nx = fmaf(th[0], x, fmaf(th[1], y, th[2]));
    const float ny = fmaf(th[3], x, fmaf(th[4], y, th[5]));

    newx[i]  = nx;
    newy[i]  = ny;
    anorm[i] = fmaf(nx, nx, ny*ny);
    bnorm[i] = fmaf(x,  x,  y*y);
}

// sorted-ascending 5-entry insertion (named registers, rare-taken branch)
#define INSERT5(dv, jv) do {                                         \
    float _d = (dv); int _j = (jv);                                  \
    if (_d < bd4) {                                                  \
        if (_d < bd3) { bd4 = bd3; bi4 = bi3;                        \
            if (_d < bd2) { bd3 = bd2; bi3 = bi2;                    \
                if (_d < bd1) { bd2 = bd1; bi2 = bi1;                \
                    if (_d < bd0) { bd1 = bd0; bi1 = bi0;            \
                                    bd0 = _d; bi0 = _j; }            \
                    else { bd1 = _d; bi1 = _j; }                     \
                } else { bd2 = _d; bi2 = _j; }                       \
            } else { bd3 = _d; bi3 = _j; }                           \
        } else { bd4 = _d; bi4 = _j; }                               \
    }                                                                \
} while (0)

// ---------------------------------------------------------------------------
// Kernel 2: fused distance-tile (WMMA f32 16x16x4) -> kNN top-5 -> weighted
// property blend. One block per 16-row tile; 8 waves split the 8192 columns.
//
// WMMA trick: sq(m,n) = ||a_m||^2 + ||b_n||^2 - 2 a_m.b_n as a K=4 dot:
//   A row m = (-2ax, -2ay, ||a||^2, 1)   B col n = (bx, by, 1, ||b||^2)
// A layout (16x4 f32): lanes 0-15 carry K=0,1 ; lanes 16-31 carry K=2,3.
//
// Top-k runs on SQUARED distances (sqrt is monotone); sqrt applied only to
// the final 5 values per row.
// ---------------------------------------------------------------------------
__global__ void st_knn_kernel(const float* __restrict__ pos,
                              const float* __restrict__ props,
                              const float* __restrict__ newx,
                              const float* __restrict__ newy,
                              const float* __restrict__ anorm,
                              const float* __restrict__ bnorm,
                              float* __restrict__ out)
{
    __shared__ float scx[STAGE], scy[STAGE], scn[STAGE];
    __shared__ __align__(16) float sdist[WAVES][16][CPAD];
    __shared__ float smd[WAVES][16][KNN];
    __shared__ int   smi[WAVES][16][KNN];

    const int t    = threadIdx.x;
    const int wave = t >> 5;
    const int lane = t & 31;
    const int half = lane >> 4;   // 0: K=0,1 lanes / cols 0-31 ; 1: K=2,3 / cols 32-63
    const int m    = lane & 15;   // row within tile (and column within 16-col tile)

    const int rowBase = blockIdx.x * 16;
    const int row     = rowBase + m;

    // A matrix operand (constant over all column chunks)
    const float ax = newx[row], ay = newy[row], an = anorm[row];
    v2f A;
    A.x = half ? an   : (-2.0f * ax);
    A.y = half ? 1.0f : (-2.0f * ay);

    float bd0 = 1e30f, bd1 = 1e30f, bd2 = 1e30f, bd3 = 1e30f, bd4 = 1e30f;
    int   bi0 = 0,     bi1 = 0,     bi2 = 0,     bi3 = 0,     bi4 = 0;

    for (int iterBase = 0; iterBase < NPTS; iterBase += STAGE) {
        __syncthreads();  // staging buffer free to overwrite
        for (int k = t; k < STAGE; k += blockDim.x) {
            const int c = iterBase + k;
            scx[k] = pos[2*c + 0];
            scy[k] = pos[2*c + 1];
            scn[k] = bnorm[c];
        }
        __syncthreads();  // staging loaded

        const int waveCol = wave * CHUNK;            // this wave's 64-col window
        const int gBase   = iterBase + waveCol;      // global column base

#pragma unroll
        for (int tile = 0; tile < CHUNK / 16; ++tile) {
            const int cc = waveCol + tile * 16 + m;
            const float bx = scx[cc], by = scy[cc], bn = scn[cc];
            v2f B;
            B.x = half ? 1.0f : bx;
            B.y = half ? bn   : by;

            v8f C = {};
            v8f D = __builtin_amdgcn_wmma_f32_16x16x4_f32(
                        false, A, false, B, (short)0, C, false, false);

            // store raw squared distances (no per-element sqrt!)
#pragma unroll
            for (int v = 0; v < 8; ++v) {
                sdist[wave][v + half * 8][tile * 16 + m] = D[v];  // lanes>=16 -> M+8
            }
        }
        __syncthreads();  // dist tile visible

        // scan: 2 lanes per row, 32 columns each via 8x ds_load_b128,
        // maintain sorted top-5 on squared distance
        const float4* rowp = (const float4*)(&sdist[wave][m][half * 32]);
        const int     jb   = gBase + half * 32;
#pragma unroll
        for (int q = 0; q < 8; ++q) {
            const float4 d4 = rowp[q];
            const int j = jb + q * 4;
            INSERT5(d4.x, j + 0);
            INSERT5(d4.y, j + 1);
            INSERT5(d4.z, j + 2);
            INSERT5(d4.w, j + 3);
        }
    }

    // merge lane pairs (lane r <-> lane r+16 hold the same row)
    {
        const float od0 = __shfl_xor(bd0, 16, 32); const int oi0 = __shfl_xor(bi0, 16, 32);
        const float od1 = __shfl_xor(bd1, 16, 32); const int oi1 = __shfl_xor(bi1, 16, 32);
        const float od2 = __shfl_xor(bd2, 16, 32); const int oi2 = __shfl_xor(bi2, 16, 32);
        const float od3 = __shfl_xor(bd3, 16, 32); const int oi3 = __shfl_xor(bi3, 16, 32);
        const float od4 = __shfl_xor(bd4, 16, 32); const int oi4 = __shfl_xor(bi4, 16, 32);
        INSERT5(od0, oi0); INSERT5(od1, oi1); INSERT5(od2, oi2);
        INSERT5(od3, oi3); INSERT5(od4, oi4);
    }

    if (half == 0) {
        smd[wave][m][0] = bd0; smi[wave][m][0] = bi0;
        smd[wave][m][1] = bd1; smi[wave][m][1] = bi1;
        smd[wave][m][2] = bd2; smi[wave][m][2] = bi2;
        smd[wave][m][3] = bd3; smi[wave][m][3] = bi3;
        smd[wave][m][4] = bd4; smi[wave][m][4] = bi4;
    }
    __syncthreads();

    // 8-way merge + sqrt + weighted blend: thread t < 16 owns row t
    if (t < 16) {
        const int r = rowBase + t;
        float bd0 = smd[0][t][0], bd1 = smd[0][t][1], bd2 = smd[0][t][2],
              bd3 = smd[0][t][3], bd4 = smd[0][t][4];
        int   bi0 = smi[0][t][0], bi1 = smi[0][t][1], bi2 = smi[0][t][2],
              bi3 = smi[0][t][3], bi4 = smi[0][t][4];
        for (int w = 1; w < WAVES; ++w) {
#pragma unroll
            for (int k = 0; k < KNN; ++k) {
                INSERT5(smd[w][t][k], smi[w][t][k]);
            }
        }
        // deferred sqrt: only 5 values per row, single-instruction v_sqrt_f32
        const float d0 = __builtin_amdgcn_sqrtf(fmaxf(bd0, 1e-12f));
        const float d1 = __builtin_amdgcn_sqrtf(fmaxf(bd1, 1e-12f));
        const float d2 = __builtin_amdgcn_sqrtf(fmaxf(bd2, 1e-12f));
        const float d3 = __builtin_amdgcn_sqrtf(fmaxf(bd3, 1e-12f));
        const float d4 = __builtin_amdgcn_sqrtf(fmaxf(bd4, 1e-12f));
        const float w0 = 1.0f / (d0 + EPSW);
        const float w1 = 1.0f / (d1 + EPSW);
        const float w2 = 1.0f / (d2 + EPSW);
        const float w3 = 1.0f / (d3 + EPSW);
        const float w4 = 1.0f / (d4 + EPSW);
        const float inv = 1.0f / (w0 + w1 + w2 + w3 + w4);
        const float* p0 = props + 3*bi0;
        const float* p1 = props + 3*bi1;
        const float* p2 = props + 3*bi2;
        const float* p3 = props + 3*bi3;
        const float* p4 = props + 3*bi4;
#pragma unroll
        for (int c = 0; c < 3; ++c) {
            const float acc = w0*p0[c] + w1*p1[c] + w2*p2[c] + w3*p3[c] + w4*p4[c];
            out[3*r + c] = acc * inv;
        }
    }
}

extern "C" void kernel_launch(void* const* d_in, const int* in_sizes, int n_in,
                              void* d_out, int out_size, void* d_ws, size_t ws_size,
                              hipStream_t stream) {
    const float* pos   = (const float*)d_in[0];  // [N,2]
    const float* props = (const float*)d_in[1];  // [N,3]
    const float* W1    = (const float*)d_in[2];
    const float* b1    = (const float*)d_in[3];
    const float* W2    = (const float*)d_in[4];
    const float* b2    = (const float*)d_in[5];
    const float* W3    = (const float*)d_in[6];
    const float* b3    = (const float*)d_in[7];
    float* out = (float*)d_out;

    float* newx  = (float*)d_ws;
    float* newy  = newx  + NPTS;
    float* anorm = newy  + NPTS;
    float* bnorm = anorm + NPTS;

    st_mlp_kernel<<<NPTS / 256, 256, 0, stream>>>(
        pos, W1, b1, W2, b2, W3, b3, newx, newy, anorm, bnorm);

    st_knn_kernel<<<NPTS / 16, 256, 0, stream>>>(
        pos, props, newx, newy, anorm, bnorm, out);
}